// VecIntInitializer_44813688766543
// MI455X (gfx1250) — compile-verified
//
#include <hip/hip_runtime.h>
#include <math.h>

// ---------------------------------------------------------------------------
// Problem constants (from the reference)
// ---------------------------------------------------------------------------
#define B_N  2048
#define TH   8
#define KM   20
#define TFUT 40
#define R_N  (B_N * KM)   // 40960 rows for the m2m stage

typedef float v2f __attribute__((ext_vector_type(2)));
typedef float v8f __attribute__((ext_vector_type(8)));

// ---------------------------------------------------------------------------
// Generic fp32 WMMA GEMM:
//   C[m][n] = act( alpha * sum_k A[m][k] * B(k,n)  + bias[n] + res[m][n] )
// transB==1: B(k,n) = Bm[n*ldb + k]   (weights stored (N,K), i.e. A @ W^T)
// transB==0: B(k,n) = Bm[k*ldb + n]   (plain A @ B)
//
// Each wave computes a 16x64 output tile = 4 accumulators sharing one A
// operand (4 WMMAs per A fetch). The k-loop is software-pipelined: operands
// for step i+1 are loaded before the 4 WMMAs of step i are issued, so the
// s_wait_loadcnt before the WMMA group overlaps with outstanding loads
// instead of draining to zero every step.
//
// V_WMMA_F32_16X16X4_F32 lane layouts:
//   A (16x4): lane(half h, r): a = A[row0+r][k0+2h .. +1]
//   B (4x16): lane(half h, r): b = B[k0+2h .. +1][col0+r]
//   C/D     : lane(h, r):      acc[g] = C[row0 + g + 8h][col0 + r]
// ---------------------------------------------------------------------------
__global__ __launch_bounds__(128) void gemm_f32_wmma(
    const float* __restrict__ A, long lda,
    const float* __restrict__ Bm, long ldb, int transB,
    const float* __restrict__ bias,
    const float* __restrict__ res, long ldres,
    float* __restrict__ C, long ldc,
    int M, int N, int K, float alpha, int act)
{
    const int wave = threadIdx.x >> 5;
    const int lane = threadIdx.x & 31;
    const int half = lane >> 4;
    const int r    = lane & 15;
    const int row0 = blockIdx.x * 64 + wave * 16;
    const int col0 = blockIdx.y * 64;
    if (row0 >= M || col0 >= N) return;   // wave-uniform exit

    const int arow = min(row0 + r, M - 1);          // clamp to stay in-bounds
    const float* __restrict__ Arow = A + (long)arow * lda;

    int bcol[4];
    const float* Brow[4];
#pragma unroll
    for (int j = 0; j < 4; ++j) {
        bcol[j] = min(col0 + j * 16 + r, N - 1);
        Brow[j] = Bm + (long)bcol[j] * ldb;         // used when transB == 1
    }

    v8f acc[4] = {};
    const int kfull = K & ~3;
    if (transB) {
        if (kfull > 0) {
            // prologue: fetch step 0 operands
            v2f a0, b0[4];
            {
                const int ka = 2 * half;
                a0.x = Arow[ka]; a0.y = Arow[ka + 1];
#pragma unroll
                for (int j = 0; j < 4; ++j) { b0[j].x = Brow[j][ka]; b0[j].y = Brow[j][ka + 1]; }
            }
            for (int k0 = 4; k0 < kfull; k0 += 4) {
                // prefetch next step before consuming current operands
                const int kn = k0 + 2 * half;
                v2f a1, b1[4];
                a1.x = Arow[kn]; a1.y = Arow[kn + 1];
#pragma unroll
                for (int j = 0; j < 4; ++j) { b1[j].x = Brow[j][kn]; b1[j].y = Brow[j][kn + 1]; }
#pragma unroll
                for (int j = 0; j < 4; ++j)
                    acc[j] = __builtin_amdgcn_wmma_f32_16x16x4_f32(
                                 false, a0, false, b0[j], (short)0, acc[j], false, false);
                a0 = a1;
#pragma unroll
                for (int j = 0; j < 4; ++j) b0[j] = b1[j];
            }
#pragma unroll
            for (int j = 0; j < 4; ++j)
                acc[j] = __builtin_amdgcn_wmma_f32_16x16x4_f32(
                             false, a0, false, b0[j], (short)0, acc[j], false, false);
        }
        if (kfull < K) {
            const int ka = kfull + 2 * half;
            v2f a;
            a.x = (ka     < K) ? Arow[ka]     : 0.0f;
            a.y = (ka + 1 < K) ? Arow[ka + 1] : 0.0f;
#pragma unroll
            for (int j = 0; j < 4; ++j) {
                v2f b;
                b.x = (ka     < K) ? Brow[j][ka]     : 0.0f;
                b.y = (ka + 1 < K) ? Brow[j][ka + 1] : 0.0f;
                acc[j] = __builtin_amdgcn_wmma_f32_16x16x4_f32(
                             false, a, false, b, (short)0, acc[j], false, false);
            }
        }
    } else {
        if (kfull > 0) {
            v2f a0, b0[4];
            {
                const int ka = 2 * half;
                a0.x = Arow[ka]; a0.y = Arow[ka + 1];
                const float* __restrict__ B0 = Bm + (long)ka * ldb;
                const float* __restrict__ B1 = Bm + (long)(ka + 1) * ldb;
#pragma unroll
                for (int j = 0; j < 4; ++j) { b0[j].x = B0[bcol[j]]; b0[j].y = B1[bcol[j]]; }
            }
            for (int k0 = 4; k0 < kfull; k0 += 4) {
                const int kn = k0 + 2 * half;
                v2f a1, b1[4];
                a1.x = Arow[kn]; a1.y = Arow[kn + 1];
                const float* __restrict__ B0 = Bm + (long)kn * ldb;
                const float* __restrict__ B1 = Bm + (long)(kn + 1) * ldb;
#pragma unroll
                for (int j = 0; j < 4; ++j) { b1[j].x = B0[bcol[j]]; b1[j].y = B1[bcol[j]]; }
#pragma unroll
                for (int j = 0; j < 4; ++j)
                    acc[j] = __builtin_amdgcn_wmma_f32_16x16x4_f32(
                                 false, a0, false, b0[j], (short)0, acc[j], false, false);
                a0 = a1;
#pragma unroll
                for (int j = 0; j < 4; ++j) b0[j] = b1[j];
            }
#pragma unroll
            for (int j = 0; j < 4; ++j)
                acc[j] = __builtin_amdgcn_wmma_f32_16x16x4_f32(
                             false, a0, false, b0[j], (short)0, acc[j], false, false);
        }
        if (kfull < K) {
            const int ka = kfull + 2 * half;
            v2f a;
            a.x = (ka     < K) ? Arow[ka]     : 0.0f;
            a.y = (ka + 1 < K) ? Arow[ka + 1] : 0.0f;
#pragma unroll
            for (int j = 0; j < 4; ++j) {
                v2f b;
                b.x = (ka     < K) ? Bm[(long)ka * ldb + bcol[j]]       : 0.0f;
                b.y = (ka + 1 < K) ? Bm[(long)(ka + 1) * ldb + bcol[j]] : 0.0f;
                acc[j] = __builtin_amdgcn_wmma_f32_16x16x4_f32(
                             false, a, false, b, (short)0, acc[j], false, false);
            }
        }
    }

#pragma unroll
    for (int j = 0; j < 4; ++j) {
        const int col = col0 + j * 16 + r;
        if (col < N) {
            const float bv = bias ? bias[col] : 0.0f;
#pragma unroll
            for (int g = 0; g < 8; ++g) {
                const int m = row0 + g + 8 * half;
                if (m < M) {
                    float v = acc[j][g] * alpha + bv;
                    if (res) v += res[(long)m * ldres + col];
                    if (act == 1) v = fmaxf(v, 0.0f);
                    C[(long)m * ldc + col] = v;
                }
            }
        }
    }
}

// ---------------------------------------------------------------------------
// Elementwise / reduction kernels
// ---------------------------------------------------------------------------
__global__ void k_maskf(const int* __restrict__ mask, float* __restrict__ mf, long n) {
    long i = blockIdx.x * (long)blockDim.x + threadIdx.x;
    if (i >= n) return;
    int m = mask[i];
    mf[i] = (m == 0) ? -INFINITY : ((m == 1) ? 0.0f : (float)m);
}

__global__ void k_add(const float* __restrict__ a, const float* __restrict__ b,
                      float* __restrict__ o, long n) {
    long i = blockIdx.x * (long)blockDim.x + threadIdx.x;
    if (i >= n) return;
    o[i] = a[i] + b[i];
}

// softmax over each row of length L; one 256-thread block per row
__global__ void k_softmax_rows(float* __restrict__ X, int L) {
    long row = blockIdx.x;
    float* x = X + row * (long)L;
    __shared__ float red[256];
    int tid = threadIdx.x;
    float mx = -INFINITY;
    for (int i = tid; i < L; i += 256) mx = fmaxf(mx, x[i]);
    red[tid] = mx; __syncthreads();
    for (int s = 128; s > 0; s >>= 1) { if (tid < s) red[tid] = fmaxf(red[tid], red[tid + s]); __syncthreads(); }
    mx = red[0]; __syncthreads();
    float sm = 0.0f;
    for (int i = tid; i < L; i += 256) { float e = expf(x[i] - mx); x[i] = e; sm += e; }
    red[tid] = sm; __syncthreads();
    for (int s = 128; s > 0; s >>= 1) { if (tid < s) red[tid] += red[tid + s]; __syncthreads(); }
    float inv = 1.0f / red[0];
    for (int i = tid; i < L; i += 256) x[i] *= inv;
}

// O[row] = LN(X[row] + Y[row]) * g + b ; D <= 256 ; blockDim must be 256
__global__ void k_add_ln(const float* __restrict__ X, const float* __restrict__ Y,
                         const float* __restrict__ g, const float* __restrict__ bta,
                         float* __restrict__ O, int D) {
    long row = blockIdx.x;
    int tid = threadIdx.x;
    __shared__ float red[256];
    float v = 0.0f;
    if (tid < D) v = X[row * (long)D + tid] + Y[row * (long)D + tid];
    red[tid] = (tid < D) ? v : 0.0f; __syncthreads();
    for (int s = 128; s > 0; s >>= 1) { if (tid < s) red[tid] += red[tid + s]; __syncthreads(); }
    float mean = red[0] / (float)D; __syncthreads();
    float d = (tid < D) ? (v - mean) : 0.0f;
    red[tid] = d * d; __syncthreads();
    for (int s = 128; s > 0; s >>= 1) { if (tid < s) red[tid] += red[tid + s]; __syncthreads(); }
    float var = red[0] / (float)D;
    float rstd = rsqrtf(var + 1e-5f);
    if (tid < D) O[row * (long)D + tid] = d * rstd * g[tid] + bta[tid];
}

// 1D conv (pad 1, kernel 3) + bias + relu: X (B,T,cin) -> Y (T,B,32)
__global__ void k_conv(const float* __restrict__ X, const float* __restrict__ W,
                       const float* __restrict__ bc, float* __restrict__ Y,
                       long Bn, int T, int cin) {
    long i = blockIdx.x * (long)blockDim.x + threadIdx.x;
    long n = Bn * (long)T * 32;
    if (i >= n) return;
    int o = (int)(i & 31);
    long bt = i >> 5;
    int t = (int)(bt % T);
    long b = bt / T;
    float s = bc[o];
    for (int kk = 0; kk < 3; ++kk) {
        int tt = t + kk - 1;
        if (tt >= 0 && tt < T)
            for (int ci = 0; ci < cin; ++ci)
                s += X[(b * T + tt) * cin + ci] * W[(o * cin + ci) * 3 + kk];
    }
    Y[((long)t * Bn + b) * 32 + o] = fmaxf(s, 0.0f);
}

// GRU cell: h <- (1-z)*n + z*h , gi/gh (B,768), h (B,256)
__global__ void k_gru(const float* __restrict__ gi, const float* __restrict__ gh,
                      float* __restrict__ h, long Bn) {
    long i = blockIdx.x * (long)blockDim.x + threadIdx.x;
    if (i >= Bn * 256) return;
    long b = i >> 8;
    int j = (int)(i & 255);
    float ir = gi[b * 768 + j],       hr = gh[b * 768 + j];
    float iz = gi[b * 768 + 256 + j], hz = gh[b * 768 + 256 + j];
    float in = gi[b * 768 + 512 + j], hn = gh[b * 768 + 512 + j];
    float r = 1.0f / (1.0f + expf(-(ir + hr)));
    float z = 1.0f / (1.0f + expf(-(iz + hz)));
    float n = tanhf(in + r * hn);
    h[i] = (1.0f - z) * n + z * h[i];
}

__global__ void k_concat2(const float* __restrict__ a, int wa,
                          const float* __restrict__ b, int wb,
                          float* __restrict__ o, long rows) {
    int w = wa + wb;
    long n = rows * (long)w;
    long i = blockIdx.x * (long)blockDim.x + threadIdx.x;
    if (i >= n) return;
    long rr = i / w; int c = (int)(i % w);
    o[i] = (c < wa) ? a[rr * wa + c] : b[rr * wb + (c - wa)];
}

__global__ void k_concat3(const float* __restrict__ a, int wa,
                          const float* __restrict__ b, int wb,
                          const float* __restrict__ c3, int wc,
                          float* __restrict__ o, long rows) {
    int w = wa + wb + wc;
    long n = rows * (long)w;
    long i = blockIdx.x * (long)blockDim.x + threadIdx.x;
    if (i >= n) return;
    long rr = i / w; int c = (int)(i % w);
    if (c < wa)           o[i] = a[rr * wa + c];
    else if (c < wa + wb) o[i] = b[rr * wb + (c - wa)];
    else                  o[i] = c3[rr * wc + (c - wa - wb)];
}

// _future: gv (R,T,2), gi (R,T,3), gs (R,T,1) -> out (R,T,3)
__global__ void k_future(const float* __restrict__ gv, const float* __restrict__ gi,
                         const float* __restrict__ gs, float* __restrict__ out,
                         long R, int T) {
    long i = blockIdx.x * (long)blockDim.x + threadIdx.x;
    if (i >= R * (long)T) return;
    long rr = i / T; int t = (int)(i % T);
    const float* gvr = gv + rr * (long)T * 2;
    float ox, oy, oz;
    if (t == 0) { ox = gvr[0]; oy = gvr[1]; oz = 0.0f; }
    else {
        float ax = gvr[(t - 1) * 2], ay = gvr[(t - 1) * 2 + 1], az = 0.0f;
        const float* c = gi + (rr * (long)T + t) * 3;
        float cx = c[0], cy = c[1], cz = c[2];
        float d = gs[rr * (long)T + t];
        float an = sqrtf(ax * ax + ay * ay + az * az);
        float cn2 = cx * cx + cy * cy + cz * cz;
        float bn = sqrtf(cn2 + d * d) / an;
        float bx = ay * cz - az * cy;
        float by = az * cx - ax * cz;
        float bz = ax * cy - ay * cx;
        float inv = 1.0f / sqrtf(bx * bx + by * by + bz * bz);
        ox = bn * bx * inv; oy = bn * by * inv; oz = bn * bz * inv;
    }
    out[i * 3 + 0] = ox; out[i * 3 + 1] = oy; out[i * 3 + 2] = oz;
}

__global__ void k_copy_xy(const float* __restrict__ in3, float* __restrict__ out2, long n) {
    long i = blockIdx.x * (long)blockDim.x + threadIdx.x;
    if (i >= n) return;
    out2[i * 2 + 0] = in3[i * 3 + 0];
    out2[i * 2 + 1] = in3[i * 3 + 1];
}

// small per-batch attention for m2m: 20 tokens, 2 heads, dh=64, qkv ld 384
__global__ __launch_bounds__(128) void k_attn_small(const float* __restrict__ qkv,
                                                    float* __restrict__ out) {
    __shared__ float sq[20 * 64], sk[20 * 64], sv[20 * 64], ss[20 * 20];
    const int b = blockIdx.x, hh = blockIdx.y, tid = threadIdx.x;
    for (int idx = tid; idx < 20 * 64; idx += 128) {
        int t = idx / 64, d = idx % 64;
        long base = ((long)b * 20 + t) * 384 + hh * 64;
        sq[idx] = qkv[base + d];
        sk[idx] = qkv[base + 128 + d];
        sv[idx] = qkv[base + 256 + d];
    }
    __syncthreads();
    for (int idx = tid; idx < 400; idx += 128) {
        int i = idx / 20, j = idx % 20;
        float s = 0.0f;
        for (int d = 0; d < 64; ++d) s += sq[i * 64 + d] * sk[j * 64 + d];
        ss[idx] = s * 0.125f;   // 1/sqrt(64)
    }
    __syncthreads();
    if (tid < 20) {
        float mx = -INFINITY;
        for (int j = 0; j < 20; ++j) mx = fmaxf(mx, ss[tid * 20 + j]);
        float sm = 0.0f;
        for (int j = 0; j < 20; ++j) { float e = expf(ss[tid * 20 + j] - mx); ss[tid * 20 + j] = e; sm += e; }
        float inv = 1.0f / sm;
        for (int j = 0; j < 20; ++j) ss[tid * 20 + j] *= inv;
    }
    __syncthreads();
    for (int idx = tid; idx < 20 * 64; idx += 128) {
        int i = idx / 64, d = idx % 64;
        float o = 0.0f;
        for (int j = 0; j < 20; ++j) o += ss[i * 20 + j] * sv[j * 64 + d];
        out[((long)b * 20 + i) * 128 + hh * 64 + d] = o;
    }
}

// softmax over K=20 + argmax (first max), one thread per batch row
__global__ void k_prob(const float* __restrict__ logits, float* __restrict__ prob,
                       int* __restrict__ best, long Bn, int Kt) {
    long b = blockIdx.x * (long)blockDim.x + threadIdx.x;
    if (b >= Bn) return;
    const float* l = logits + b * Kt;
    float mx = l[0]; int am = 0;
    for (int i = 1; i < Kt; ++i) if (l[i] > mx) { mx = l[i]; am = i; }
    float s = 0.0f;
    for (int i = 0; i < Kt; ++i) s += expf(l[i] - mx);
    float inv = 1.0f / s;
    for (int i = 0; i < Kt; ++i) prob[b * Kt + i] = expf(l[i] - mx) * inv;
    best[b] = am;
}

__global__ void k_gather(const float* __restrict__ gi, const float* __restrict__ gs,
                         const int* __restrict__ best, float* __restrict__ bi,
                         float* __restrict__ bs, long Bn, int Kt, int T) {
    long i = blockIdx.x * (long)blockDim.x + threadIdx.x;
    if (i >= Bn * (long)T) return;
    long b = i / T; int t = (int)(i % T);
    long src = ((b * Kt + best[b]) * (long)T + t);
    bi[i * 3 + 0] = gi[src * 3 + 0];
    bi[i * 3 + 1] = gi[src * 3 + 1];
    bi[i * 3 + 2] = gi[src * 3 + 2];
    bs[i] = gs[src];
}

__global__ void k_conc(const float* __restrict__ c, float* __restrict__ o, long n) {
    long i = blockIdx.x * (long)blockDim.x + threadIdx.x;
    if (i >= n) return;
    float x = c[i];
    float e = (x > 0.0f) ? x : (expf(x) - 1.0f);
    o[i] = 1.0f / (e + 1.0f + 0.02f);
}

// ---------------------------------------------------------------------------
// Host orchestration
// ---------------------------------------------------------------------------
struct Lin { const float* w; const float* b; };
struct Mlp3 { Lin l0, l1, l2; };
struct TxLayer { Lin ff1, ff2; const float *ln1g, *ln1b, *ln2g, *ln2b; Lin out, qkv; };
struct TxEnc { const float* enc_w; TxLayer l0, l1; };
struct StEnc { const float *b_hh, *b_ih, *conv_b, *conv_w, *w_hh, *w_ih; int cin; };

static inline dim3 g1d(long n, int bs = 256) { return dim3((unsigned)((n + bs - 1) / bs)); }

extern "C" void kernel_launch(void* const* d_in, const int* in_sizes, int n_in,
                              void* d_out_v, int out_size, void* d_ws, size_t ws_size,
                              hipStream_t stream) {
    (void)in_sizes; (void)n_in; (void)out_size; (void)ws_size;
    int idx = 0;
    auto ld = [&]() { return (const float*)d_in[idx++]; };

    const float* in_x    = ld();                    // (B, 8, 6)
    const float* in_int  = ld();                    // (B, 8, 3)
    const float* in_sim  = ld();                    // (B, 8, 3)
    const int*   in_mask = (const int*)d_in[idx++]; // (B, B)

    // params flattened as a JAX pytree: dict keys sorted recursively
    auto lin = [&]() { Lin L; L.w = ld(); L.b = ld(); return L; };
    auto ldst = [&](int cin) { StEnc s; s.b_hh = ld(); s.b_ih = ld(); s.conv_b = ld();
                               s.conv_w = ld(); s.w_hh = ld(); s.w_ih = ld(); s.cin = cin; return s; };
    auto ldmlp = [&]() { Mlp3 m; m.l0 = lin(); m.l1 = lin(); m.l2 = lin(); return m; };
    auto ldtxl = [&]() { TxLayer t; t.ff1 = lin(); t.ff2 = lin();
                         t.ln1g = ld(); t.ln1b = ld(); t.ln2g = ld(); t.ln2b = ld();
                         t.out = lin(); t.qkv = lin(); return t; };
    auto ldtx = [&]() { TxEnc e; e.enc_w = ld(); e.l0 = ldtxl(); e.l1 = ldtxl(); return e; };

    StEnc  p_ego_int   = ldst(3);
    StEnc  p_ego_mean  = ldst(6);
    StEnc  p_ego_scale = ldst(6);
    StEnc  p_ego_sim   = ldst(3);
    StEnc  p_ego_var   = ldst(6);
    Mlp3   p_int_dec   = ldmlp();
    TxEnc  p_int_soc   = ldtx();
    Lin    p_m2m_in    = lin();
    TxLayer p_m2m      = ldtxl();
    Mlp3   p_mean_dec  = ldmlp();
    Mlp3   p_scale_dec = ldmlp();
    Mlp3   p_scale_enc = ldmlp();
    Mlp3   p_sim_dec   = ldmlp();
    TxEnc  p_sim_soc   = ldtx();
    TxEnc  p_soc       = ldtx();
    Mlp3   p_to_conc   = ldmlp();
    Mlp3   p_to_prob   = ldmlp();
    Mlp3   p_var_dec   = ldmlp();

    float* d_out = (float*)d_out_v;
    // output offsets (flat concat of the returned tuple, fp32)
    const size_t OUT_GVAR = 0;                                  // (B,K,40,2)
    const size_t OUT_GMEAN = OUT_GVAR + (size_t)B_N * 1600;     // (B,40,2)
    const size_t OUT_GSC   = OUT_GMEAN + (size_t)B_N * 80;      // (B,1)
    const size_t OUT_GINT  = OUT_GSC + B_N;                     // (B,K,40,3)
    const size_t OUT_GSIM  = OUT_GINT + (size_t)B_N * 2400;     // (B,K,40,1)
    const size_t OUT_GOALV = OUT_GSIM + (size_t)B_N * 800;      // (B,K,40,2)
    const size_t OUT_GOALM = OUT_GOALV + (size_t)B_N * 1600;    // (B,40,2)
    const size_t OUT_CONC  = OUT_GOALM + (size_t)B_N * 80;      // (B,K,40,1)
    const size_t OUT_PROB  = OUT_CONC + (size_t)B_N * 800;      // (B,K)

    // workspace arena (floats)
    float* W = (float*)d_ws;
    size_t off = 0;
    auto arena = [&](size_t n) { float* p = W + off; off += n; return p; };
    // persistent
    float* soc_out[3]; for (int i = 0; i < 3; ++i) soc_out[i] = arena((size_t)B_N * 256);
    float* h_st[5];    for (int i = 0; i < 5; ++i) h_st[i]   = arena((size_t)B_N * 256);
    float* goal3 = arena((size_t)R_N * TFUT * 3);
    int*   best  = (int*)arena(B_N);
    float* bi    = arena((size_t)B_N * TFUT * 3);
    float* bs    = arena((size_t)B_N * TFUT);
    const size_t scratch0 = off;

    auto gemm = [&](const float* A, long lda, const float* Bm, long ldb, int tB,
                    const float* bias, const float* res, long ldres,
                    float* C, long ldc, int M, int N, int K,
                    float alpha = 1.0f, int act = 0) {
        dim3 grid((M + 63) / 64, (N + 63) / 64);
        gemm_f32_wmma<<<grid, 128, 0, stream>>>(A, lda, Bm, ldb, tB, bias, res, ldres,
                                                C, ldc, M, N, K, alpha, act);
    };
    // relu on all but the last layer
    auto mlp3 = [&](const float* x, int din, const Mlp3& P, int d0, int d1, int dout,
                    long rows, float* h0, float* h1, float* outp) {
        gemm(x, din, P.l0.w, din, 1, P.l0.b, nullptr, 0, h0, d0, (int)rows, d0, din, 1.0f, 1);
        gemm(h0, d0, P.l1.w, d0, 1, P.l1.b, nullptr, 0, h1, d1, (int)rows, d1, d0, 1.0f, 1);
        gemm(h1, d1, P.l2.w, d1, 1, P.l2.b, nullptr, 0, outp, dout, (int)rows, dout, d1, 1.0f, 0);
    };

    // ---------------- Phase 1: social encoders -----------------------------
    off = scratch0;
    float* mf     = arena((size_t)B_N * B_N);
    float* scores = arena((size_t)2 * B_N * B_N);
    float* qkv    = arena((size_t)B_N * 768);
    float* fbuf   = arena((size_t)B_N * 256);
    float* zbuf   = arena((size_t)B_N * 256);
    float* attno  = arena((size_t)B_N * 256);
    float* tA     = arena((size_t)B_N * 256);
    float* tB     = arena((size_t)B_N * 256);

    k_maskf<<<g1d((long)B_N * B_N), 256, 0, stream>>>(in_mask, mf, (long)B_N * B_N);

    const float inv_sqrt_dh = 1.0f / sqrtf(128.0f);
    auto social = [&](const float* xin, int in_dim, const TxEnc& P, float* outp) {
        gemm(xin, in_dim, P.enc_w, in_dim, 1, nullptr, nullptr, 0, fbuf, 256, B_N, 256, in_dim);
        hipMemcpyAsync(zbuf, fbuf, (size_t)B_N * 256 * sizeof(float),
                       hipMemcpyDeviceToDevice, stream);
        const TxLayer* Ls[2] = { &P.l0, &P.l1 };
        for (int li = 0; li < 2; ++li) {
            const TxLayer& L = *Ls[li];
            gemm(zbuf, 256, L.qkv.w, 256, 1, L.qkv.b, nullptr, 0, qkv, 768, B_N, 768, 256);
            for (int h = 0; h < 2; ++h)
                gemm(qkv + h * 128, 768, qkv + 256 + h * 128, 768, 1,
                     nullptr, mf, B_N, scores + (size_t)h * B_N * B_N, B_N,
                     B_N, B_N, 128, inv_sqrt_dh, 0);
            k_softmax_rows<<<2 * B_N, 256, 0, stream>>>(scores, B_N);
            for (int h = 0; h < 2; ++h)
                gemm(scores + (size_t)h * B_N * B_N, B_N, qkv + 512 + h * 128, 768, 0,
                     nullptr, nullptr, 0, attno + h * 128, 256, B_N, 128, B_N);
            gemm(attno, 256, L.out.w, 256, 1, L.out.b, nullptr, 0, tA, 256, B_N, 256, 256);
            k_add_ln<<<B_N, 256, 0, stream>>>(zbuf, tA, L.ln1g, L.ln1b, zbuf, 256);
            gemm(zbuf, 256, L.ff1.w, 256, 1, L.ff1.b, nullptr, 0, tA, 256, B_N, 256, 256, 1.0f, 1);
            gemm(tA, 256, L.ff2.w, 256, 1, L.ff2.b, nullptr, 0, tB, 256, B_N, 256, 256);
            k_add_ln<<<B_N, 256, 0, stream>>>(zbuf, tB, L.ln2g, L.ln2b, zbuf, 256);
        }
        k_add<<<g1d((long)B_N * 256), 256, 0, stream>>>(fbuf, zbuf, outp, (long)B_N * 256);
    };
    social(in_x,   48, p_soc,     soc_out[0]);
    social(in_int, 24, p_int_soc, soc_out[1]);
    social(in_sim, 24, p_sim_soc, soc_out[2]);

    // ---------------- Phase 2: conv + GRU encoders --------------------------
    off = scratch0;
    float* ybuf = arena((size_t)TH * B_N * 32);
    float* gibuf = arena((size_t)B_N * 768);
    float* ghbuf = arena((size_t)B_N * 768);
    auto st = [&](const float* Xin, const StEnc& P, float* hT) {
        k_conv<<<g1d((long)B_N * TH * 32), 256, 0, stream>>>(Xin, P.conv_w, P.conv_b,
                                                             ybuf, B_N, TH, P.cin);
        hipMemsetAsync(hT, 0, (size_t)B_N * 256 * sizeof(float), stream);
        for (int t = 0; t < TH; ++t) {
            gemm(ybuf + (size_t)t * B_N * 32, 32, P.w_ih, 32, 1, P.b_ih, nullptr, 0,
                 gibuf, 768, B_N, 768, 32);
            gemm(hT, 256, P.w_hh, 256, 1, P.b_hh, nullptr, 0, ghbuf, 768, B_N, 768, 256);
            k_gru<<<g1d((long)B_N * 256), 256, 0, stream>>>(gibuf, ghbuf, hT, B_N);
        }
    };
    st(in_x,   p_ego_var,   h_st[0]);
    st(in_x,   p_ego_mean,  h_st[1]);
    st(in_x,   p_ego_scale, h_st[2]);
    st(in_int, p_ego_int,   h_st[3]);
    st(in_sim, p_ego_sim,   h_st[4]);

    // ---------------- Phase 3: decoders -------------------------------------
    off = scratch0;
    float* cat  = arena((size_t)B_N * 544);
    float* hid1 = arena((size_t)B_N * 1024);
    float* hid2 = arena((size_t)B_N * 1024);
    float* sfeat = arena((size_t)B_N * 32);

    // guess_intention = int_dec(concat(ego_int, int_soc))
    k_concat2<<<g1d((long)B_N * 512), 256, 0, stream>>>(h_st[3], 256, soc_out[1], 256, cat, B_N);
    mlp3(cat, 512, p_int_dec, 256, 128, 2400, B_N, hid1, hid2, d_out + OUT_GINT);
    // guess_similarity = sim_dec(concat(ego_sim, sim_soc))
    k_concat2<<<g1d((long)B_N * 512), 256, 0, stream>>>(h_st[4], 256, soc_out[2], 256, cat, B_N);
    mlp3(cat, 512, p_sim_dec, 256, 128, 800, B_N, hid1, hid2, d_out + OUT_GSIM);
    // guess_scale = scale_dec(concat(ego_scale, soc))
    k_concat2<<<g1d((long)B_N * 512), 256, 0, stream>>>(h_st[2], 256, soc_out[0], 256, cat, B_N);
    mlp3(cat, 512, p_scale_dec, 256, 128, 1, B_N, hid1, hid2, d_out + OUT_GSC);
    // scale_feat = scale_enc(guess_scale)
    mlp3(d_out + OUT_GSC, 1, p_scale_enc, 4, 16, 32, B_N, hid1, hid2, sfeat);
    // guess_var = var_dec(concat(ego_var, soc, scale_feat))
    k_concat3<<<g1d((long)B_N * 544), 256, 0, stream>>>(h_st[0], 256, soc_out[0], 256,
                                                        sfeat, 32, cat, B_N);
    mlp3(cat, 544, p_var_dec, 1024, 1024, 1600, B_N, hid1, hid2, d_out + OUT_GVAR);
    // guess_mean = mean_dec(concat(ego_mean, soc))
    k_concat2<<<g1d((long)B_N * 512), 256, 0, stream>>>(h_st[1], 256, soc_out[0], 256, cat, B_N);
    mlp3(cat, 512, p_mean_dec, 256, 128, 80, B_N, hid1, hid2, d_out + OUT_GMEAN);

    // ---------------- Phase 4: goal_var = future(...) -----------------------
    k_future<<<g1d((long)R_N * TFUT), 256, 0, stream>>>(d_out + OUT_GVAR, d_out + OUT_GINT,
                                                        d_out + OUT_GSIM, goal3,
                                                        (long)R_N, TFUT);
    k_copy_xy<<<g1d((long)R_N * TFUT), 256, 0, stream>>>(goal3, d_out + OUT_GOALV,
                                                         (long)R_N * TFUT);

    // ---------------- Phase 5: m2m refinement -------------------------------
    off = scratch0;
    float* hbuf   = arena((size_t)R_N * 128);
    float* qkvm   = arena((size_t)R_N * 384);
    float* attnom = arena((size_t)R_N * 128);
    float* t1     = arena((size_t)R_N * 128);
    float* t2     = arena((size_t)R_N * 128);
    float* concraw = arena((size_t)R_N * TFUT);
    float* probraw = arena((size_t)R_N);
    float* gm3    = arena((size_t)B_N * TFUT * 3);

    gemm(goal3, 120, p_m2m_in.w, 120, 1, p_m2m_in.b, nullptr, 0, hbuf, 128, R_N, 128, 120);
    gemm(hbuf, 128, p_m2m.qkv.w, 128, 1, p_m2m.qkv.b, nullptr, 0, qkvm, 384, R_N, 384, 128);
    k_attn_small<<<dim3(B_N, 2), 128, 0, stream>>>(qkvm, attnom);
    gemm(attnom, 128, p_m2m.out.w, 128, 1, p_m2m.out.b, nullptr, 0, t1, 128, R_N, 128, 128);
    k_add_ln<<<R_N, 256, 0, stream>>>(hbuf, t1, p_m2m.ln1g, p_m2m.ln1b, hbuf, 128);
    gemm(hbuf, 128, p_m2m.ff1.w, 128, 1, p_m2m.ff1.b, nullptr, 0, t1, 128, R_N, 128, 128, 1.0f, 1);
    gemm(t1, 128, p_m2m.ff2.w, 128, 1, p_m2m.ff2.b, nullptr, 0, t2, 128, R_N, 128, 128);
    k_add_ln<<<R_N, 256, 0, stream>>>(hbuf, t2, p_m2m.ln2g, p_m2m.ln2b, hbuf, 128);

    // conc = 1/(elu(to_conc(refined)) + 1.02)
    mlp3(hbuf, 128, p_to_conc, 64, 32, TFUT, R_N, t1, t2, concraw);
    k_conc<<<g1d((long)R_N * TFUT), 256, 0, stream>>>(concraw, d_out + OUT_CONC,
                                                      (long)R_N * TFUT);
    // prob = softmax(to_prob(refined)), best = argmax
    mlp3(hbuf, 128, p_to_prob, 64, 32, 1, R_N, t1, t2, probraw);
    k_prob<<<g1d(B_N), 256, 0, stream>>>(probraw, d_out + OUT_PROB, best, B_N, KM);

    // ---------------- Phase 6: goal_mean ------------------------------------
    k_gather<<<g1d((long)B_N * TFUT), 256, 0, stream>>>(d_out + OUT_GINT, d_out + OUT_GSIM,
                                                        best, bi, bs, B_N, KM, TFUT);
    k_future<<<g1d((long)B_N * TFUT), 256, 0, stream>>>(d_out + OUT_GMEAN, bi, bs, gm3,
                                                        (long)B_N, TFUT);
    k_copy_xy<<<g1d((long)B_N * TFUT), 256, 0, stream>>>(gm3, d_out + OUT_GOALM,
                                                         (long)B_N * TFUT);
}